// INT8SymmetricWeightsDecompressor_76175539962652
// MI455X (gfx1250) — compile-verified
//
#include <hip/hip_runtime.h>

// True ext-vector types so __builtin_nontemporal_* and __builtin_convertvector
// apply cleanly.
typedef int   v4i __attribute__((ext_vector_type(4)));
typedef float v4f __attribute__((ext_vector_type(4)));

#define BLOCK 256   // 8 wave32s per block
#define ITERS 8     // 8192 cols / (4 elems * 256 threads)

// ---------------------------------------------------------------------------
// Hot path: one 256-thread block per row of the 8192x8192 matrix.
//   - scale[row] is block-uniform  -> s_load_b32, zero per-element cost
//   - 8 independent 128-bit NT loads issued back-to-back (deep LOADcnt)
//   - v_cvt_f32_i32 + v_mul_f32 per element (ALU cost ~0 vs 22us HBM floor)
//   - 8 128-bit NT stores
// NT temporal hints: 512 MB streamed exactly once > 192 MB L2 -> don't rinse.
// ---------------------------------------------------------------------------
__global__ __launch_bounds__(BLOCK)
void decompress_rows_kernel(const int* __restrict__ weight,
                            const float* __restrict__ scale,
                            float* __restrict__ out) {
    const int   row = blockIdx.x;
    const float s   = scale[row];                      // uniform -> SGPR

    const int cols4 = BLOCK * ITERS;                   // 2048 int4 per row
    const v4i* __restrict__ w4 =
        reinterpret_cast<const v4i*>(weight) + (size_t)row * cols4;
    v4f* __restrict__ o4 =
        reinterpret_cast<v4f*>(out) + (size_t)row * cols4;

    // gfx1250 global_prefetch_b8: nudge the tail of this block's segment
    // toward L2 while the leading b128 loads are in flight.
    __builtin_prefetch(&w4[threadIdx.x + (ITERS - 1) * BLOCK], 0, 0);

    v4f r[ITERS];
#pragma unroll
    for (int i = 0; i < ITERS; ++i) {
        const v4i wv = __builtin_nontemporal_load(&w4[threadIdx.x + i * BLOCK]);
        r[i] = __builtin_convertvector(wv, v4f) * s;
    }
#pragma unroll
    for (int i = 0; i < ITERS; ++i) {
        __builtin_nontemporal_store(r[i], &o4[threadIdx.x + i * BLOCK]);
    }
}

// ---------------------------------------------------------------------------
// Shape-guard fallback: 2-D grid, blockIdx.y = row. No integer division —
// scale[row] stays a uniform scalar load here too.
// ---------------------------------------------------------------------------
__global__ __launch_bounds__(BLOCK)
void decompress_generic_kernel(const int* __restrict__ weight,
                               const float* __restrict__ scale,
                               float* __restrict__ out,
                               int cols) {
    const int   row = blockIdx.y;
    const float s   = scale[row];
    const size_t base = (size_t)row * (size_t)cols;

    for (int c = blockIdx.x * BLOCK + threadIdx.x; c < cols;
         c += gridDim.x * BLOCK) {
        out[base + c] = (float)weight[base + c] * s;
    }
}

extern "C" void kernel_launch(void* const* d_in, const int* in_sizes, int n_in,
                              void* d_out, int out_size, void* d_ws, size_t ws_size,
                              hipStream_t stream) {
    (void)n_in; (void)out_size; (void)d_ws; (void)ws_size;

    const int*   weight = (const int*)d_in[0];    // int8 values in int32 container
    const float* scale  = (const float*)d_in[1];  // per-row scale, OUT entries
    float*       out    = (float*)d_out;

    const long long n    = (long long)in_sizes[0];   // OUT*IN elements
    const int       rows = in_sizes[1];              // OUT (scale count)
    const int       cols = (int)(n / rows);          // IN

    if (cols == BLOCK * ITERS * 4) {
        // 8192 columns: 1 block/row, fully unrolled b128 NT stream.
        decompress_rows_kernel<<<rows, BLOCK, 0, stream>>>(weight, scale, out);
    } else {
        const int gx = (cols + BLOCK - 1) / BLOCK;
        dim3 grid(gx < 64 ? gx : 64, rows);
        decompress_generic_kernel<<<grid, BLOCK, 0, stream>>>(weight, scale,
                                                              out, cols);
    }
}